// GNNLayer_30975304138846
// MI455X (gfx1250) — compile-verified
//
#include <hip/hip_runtime.h>
#include <hip/hip_bf16.h>

// ---------------------------------------------------------------------------
// GNN layer: support = X @ W ; out[dst] += w_e * support[src] ; +bias ;
// BatchNorm(train, biased var) ; ReLU.
// X: [100000,256] f32, W: [256,128] f32. Output [100000,128] f32.
// ---------------------------------------------------------------------------

#define IN_DIM   256
#define OUT_DIM  128
#define BN_EPS   1e-5f
#define AROW     (IN_DIM + 2)   // padded LDS row stride (dwords): 258

typedef float v2f __attribute__((ext_vector_type(2)));
typedef float v8f __attribute__((ext_vector_type(8)));

// ---------------------------------------------------------------------------
// Repack W[256][128] row-major into Bp[(k/2)][n] = (W[k][n], W[k+1][n])
// so a B fragment for V_WMMA_F32_16X16X4_F32 is a single b64 load:
//   lanes 0-15  need (K=k  , K=k+1) at col n   -> Bp[k/2    ][n]
//   lanes 16-31 need (K=k+2, K=k+3) at col n   -> Bp[k/2 + 1][n]
// ---------------------------------------------------------------------------
__global__ __launch_bounds__(256) void repack_w_kernel(
    const float* __restrict__ W, float2* __restrict__ Bp) {
  int idx = blockIdx.x * 256 + threadIdx.x;          // 128 kpairs * 128 n
  if (idx >= (IN_DIM / 2) * OUT_DIM) return;
  int kp = idx >> 7;
  int n  = idx & (OUT_DIM - 1);
  float2 v;
  v.x = W[(2 * kp)     * OUT_DIM + n];
  v.y = W[(2 * kp + 1) * OUT_DIM + n];
  Bp[idx] = v;
}

// ---------------------------------------------------------------------------
// GEMM: support[nNodes][128] = X[nNodes][256] @ W.
// Block = 64 threads = 2 waves. Each wave owns one 16-row strip and all 128
// output columns: 8 accumulator tiles of V_WMMA_F32_16X16X4_F32.
// A tile (16x256 f32) staged in LDS with padded rows to kill bank conflicts.
// ---------------------------------------------------------------------------
__global__ __launch_bounds__(64) void gemm_wmma_kernel(
    const float* __restrict__ X, const float2* __restrict__ Bp,
    float* __restrict__ support, int nNodes) {
  __shared__ float ldsA[2][16 * AROW];

  const int lane = threadIdx.x & 31;
  const int wave = threadIdx.x >> 5;
  const int strip = blockIdx.x * 2 + wave;
  const int rowBase = strip * 16;
  const bool active = rowBase < nNodes;

  float* la = ldsA[wave];

  if (active) {
    // Stage 16 x 256 A tile, coalesced float4 loads (1024 float4 / 32 lanes).
    for (int i = lane; i < 16 * (IN_DIM / 4); i += 32) {
      int r  = i >> 6;            // 64 float4 per row
      int c4 = i & 63;
      int row = rowBase + r;
      if (row >= nNodes) row = nNodes - 1;   // clamp (tail safety)
      const float4* src4 = (const float4*)(X + (size_t)row * IN_DIM);
      float4 v = src4[c4];
      float* p = la + r * AROW + c4 * 4;
      p[0] = v.x; p[1] = v.y; p[2] = v.z; p[3] = v.w;
    }
  }
  __syncthreads();
  if (!active) return;

  const int half = lane >> 4;        // 0: K pair (k,k+1), 1: (k+2,k+3)
  const int mrow = lane & 15;

  v8f acc[8];
#pragma unroll
  for (int nt = 0; nt < 8; ++nt) acc[nt] = (v8f)0.0f;

#pragma unroll 2
  for (int k = 0; k < IN_DIM; k += 4) {
    // A fragment: 32-bit A 16x4 layout — lanes 0-15: K=k,k+1; 16-31: K=k+2,k+3
    const v2f a = *(const v2f*)(la + mrow * AROW + k + half * 2);
    const int kp = (k >> 1) + half;
#pragma unroll
    for (int nt = 0; nt < 8; ++nt) {
      float2 bf = Bp[kp * OUT_DIM + nt * 16 + mrow];
      v2f b; b.x = bf.x; b.y = bf.y;
      acc[nt] = __builtin_amdgcn_wmma_f32_16x16x4_f32(
          false, a, false, b, (short)0, acc[nt], false, false);
    }
  }

  // C/D 16x16 f32 layout: lanes 0-15 N=lane, M=vgpr; lanes 16-31 N=lane-16,
  // M=vgpr+8.
  float* outBase = support + (size_t)rowBase * OUT_DIM + (size_t)half * 8 * OUT_DIM;
  if (rowBase + 16 <= nNodes) {
    // Fast path: whole strip in range — straight-line b32 stores, no exec
    // mask churn (this is the common case; nNodes % 16 == 0 here).
#pragma unroll
    for (int nt = 0; nt < 8; ++nt) {
      int col = nt * 16 + mrow;
#pragma unroll
      for (int r = 0; r < 8; ++r) {
        outBase[(size_t)r * OUT_DIM + col] = acc[nt][r];
      }
    }
  } else {
    // Generic tail: per-row guard.
#pragma unroll
    for (int nt = 0; nt < 8; ++nt) {
      int col = nt * 16 + mrow;
#pragma unroll
      for (int r = 0; r < 8; ++r) {
        int row = rowBase + r + half * 8;
        if (row < nNodes)
          support[(size_t)row * OUT_DIM + col] = acc[nt][r];
      }
    }
  }
}

// ---------------------------------------------------------------------------
// SpMM scatter: out[dst[e]][c] += edge_w[e] * support[src[e]][c].
// One thread handles one edge x 4 channels (float4 gather, 4 f32 atomics).
// support & out are L2-resident (51 MB each, 192 MB L2).
// ---------------------------------------------------------------------------
__global__ __launch_bounds__(256) void spmm_kernel(
    const float* __restrict__ support, const float* __restrict__ edge_w,
    const int* __restrict__ src, const int* __restrict__ dst,
    float* __restrict__ out, int nEdges) {
  long long idx = (long long)blockIdx.x * 256 + threadIdx.x;
  long long total = (long long)nEdges * (OUT_DIM / 4);
  if (idx >= total) return;
  int e = (int)(idx >> 5);          // 32 float4 per edge
  int q = (int)(idx & 31);
  float w = edge_w[e];
  int s = src[e];
  int d = dst[e];
  const float4* sp = (const float4*)(support + (size_t)s * OUT_DIM);
  float4 v = sp[q];
  float* op = out + (size_t)d * OUT_DIM + q * 4;
  atomicAdd(op + 0, w * v.x);
  atomicAdd(op + 1, w * v.y);
  atomicAdd(op + 2, w * v.z);
  atomicAdd(op + 3, w * v.w);
}

// ---------------------------------------------------------------------------
// Per-channel sum / sum-of-squares over nodes of (agg + bias).
// 256 threads = 2 node-rows x 128 channels; block-reduce then atomics.
// ---------------------------------------------------------------------------
__global__ __launch_bounds__(256) void stats_kernel(
    const float* __restrict__ agg, const float* __restrict__ bias,
    float* __restrict__ sums, float* __restrict__ sumsq, int nNodes) {
  const int c    = threadIdx.x & (OUT_DIM - 1);
  const int rOff = threadIdx.x >> 7;   // 0 or 1
  const float b = bias[c];
  float s = 0.0f, s2 = 0.0f;
  for (int row = blockIdx.x * 2 + rOff; row < nNodes; row += gridDim.x * 2) {
    float x = agg[(size_t)row * OUT_DIM + c] + b;
    s  += x;
    s2 += x * x;
  }
  __shared__ float red[512];
  red[threadIdx.x]       = s;
  red[256 + threadIdx.x] = s2;
  __syncthreads();
  if (rOff == 0) {
    float ts  = red[c]       + red[OUT_DIM + c];
    float ts2 = red[256 + c] + red[256 + OUT_DIM + c];
    atomicAdd(&sums[c],  ts);
    atomicAdd(&sumsq[c], ts2);
  }
}

// ---------------------------------------------------------------------------
// Fold BN params: scale = gamma*rsqrt(var+eps); shift = beta + (bias-mean)*scale.
// final = relu(agg*scale + shift).
// ---------------------------------------------------------------------------
__global__ __launch_bounds__(128) void bn_params_kernel(
    const float* __restrict__ sums, const float* __restrict__ sumsq,
    const float* __restrict__ gamma, const float* __restrict__ beta,
    const float* __restrict__ bias, float* __restrict__ scale,
    float* __restrict__ shift, int nNodes) {
  int c = threadIdx.x;
  float invN = 1.0f / (float)nNodes;
  float mean = sums[c] * invN;
  float var  = sumsq[c] * invN - mean * mean;
  if (var < 0.0f) var = 0.0f;
  float sc = gamma[c] * rsqrtf(var + BN_EPS);
  scale[c] = sc;
  shift[c] = beta[c] + (bias[c] - mean) * sc;
}

// ---------------------------------------------------------------------------
// In-place normalize + ReLU on d_out (float4 vectorized).
// ---------------------------------------------------------------------------
__global__ __launch_bounds__(256) void norm_relu_kernel(
    float* __restrict__ out, const float* __restrict__ scale,
    const float* __restrict__ shift, int nNodes) {
  long long idx = (long long)blockIdx.x * 256 + threadIdx.x;  // float4 units
  long long total = (long long)nNodes * (OUT_DIM / 4);
  if (idx >= total) return;
  int c4 = (int)(idx & 31) * 4;
  float4* p = (float4*)out + idx;
  float4 v = *p;
  v.x = fmaxf(v.x * scale[c4 + 0] + shift[c4 + 0], 0.0f);
  v.y = fmaxf(v.y * scale[c4 + 1] + shift[c4 + 1], 0.0f);
  v.z = fmaxf(v.z * scale[c4 + 2] + shift[c4 + 2], 0.0f);
  v.w = fmaxf(v.w * scale[c4 + 3] + shift[c4 + 3], 0.0f);
  *p = v;
}

// ---------------------------------------------------------------------------
extern "C" void kernel_launch(void* const* d_in, const int* in_sizes, int n_in,
                              void* d_out, int out_size, void* d_ws,
                              size_t ws_size, hipStream_t stream) {
  const float* features = (const float*)d_in[0];
  const float* weight   = (const float*)d_in[1];
  const float* bias     = (const float*)d_in[2];
  const float* gamma    = (const float*)d_in[3];
  const float* beta     = (const float*)d_in[4];
  const float* edge_w   = (const float*)d_in[5];
  const int*   src      = (const int*)d_in[6];
  const int*   dst      = (const int*)d_in[7];
  float* out = (float*)d_out;

  const int nNodes = in_sizes[0] / IN_DIM;   // 100000
  const int nEdges = in_sizes[5];            // 3200000

  // Workspace layout: support | packed B | sums | sumsq | scale | shift
  char* ws = (char*)d_ws;
  const size_t supBytes = (size_t)nNodes * OUT_DIM * sizeof(float);
  float*  support = (float*)ws;
  float2* Bp      = (float2*)(ws + supBytes);
  float*  sums    = (float*)(ws + supBytes + (IN_DIM / 2) * OUT_DIM * sizeof(float2));
  float*  sumsq   = sums + OUT_DIM;
  float*  scale   = sums + 2 * OUT_DIM;
  float*  shiftv  = sums + 3 * OUT_DIM;

  // Zero aggregation target (d_out is the accumulator) and stats.
  hipMemsetAsync(d_out, 0, (size_t)nNodes * OUT_DIM * sizeof(float), stream);
  hipMemsetAsync(sums, 0, 2 * OUT_DIM * sizeof(float), stream);

  // 1) Repack W for b64 fragment loads.
  repack_w_kernel<<<((IN_DIM / 2) * OUT_DIM + 255) / 256, 256, 0, stream>>>(
      weight, Bp);

  // 2) Dense transform via fp32 WMMA.
  int strips = (nNodes + 15) / 16;                  // 6250
  gemm_wmma_kernel<<<(strips + 1) / 2, 64, 0, stream>>>(features, Bp, support,
                                                        nNodes);

  // 3) Edge scatter-add (L2-resident atomics).
  long long workE = (long long)nEdges * (OUT_DIM / 4);
  spmm_kernel<<<(int)((workE + 255) / 256), 256, 0, stream>>>(
      support, edge_w, src, dst, out, nEdges);

  // 4) BN batch statistics.
  stats_kernel<<<512, 256, 0, stream>>>(out, bias, sums, sumsq, nNodes);

  // 5) Fold params.
  bn_params_kernel<<<1, 128, 0, stream>>>(sums, sumsq, gamma, beta, bias,
                                          scale, shiftv, nNodes);

  // 6) Normalize + ReLU in place.
  long long workN = (long long)nNodes * (OUT_DIM / 4);
  norm_relu_kernel<<<(int)((workN + 255) / 256), 256, 0, stream>>>(out, scale,
                                                                   shiftv,
                                                                   nNodes);
}